// LIChI_10007273799817
// MI455X (gfx1250) — compile-verified
//
#include <hip/hip_runtime.h>

// ---------------- problem constants (from reference) ----------------
#define Hh    128
#define Ww    128
#define Pp    11           // patch size
#define Kk    16           // group size
#define Vv    32           // WIN//2
#define WINW  65           // search window
#define HREF  40
#define WREF  40
#define HP    118          // H - P + 1
#define NGRP  (4*HREF*WREF)   // 6400
#define NPIX  (4*Hh*Ww)       // 65536
#define NPATCH (Pp*Pp)        // 121
#define KPAD  128
#define E_CONST 30.25f        // n * alpha^2 * sigma^2
#define DE_CONST 151.25f      // D + E
// BM tiling: 4x4 refs per workgroup; union candidate grid 74x74; staged 84x84
#define TGRID 74
#define TROWS 84
#define NCHUNK 343            // ceil(74*74/16)
#define NS2   (4*HP*HP)       // 55696 patch-norm entries

// in-patch pixel j -> tile-local offset (row-major 84-wide tile)
#define OFF(j) ((((j)/Pp)*84) + ((j)%Pp))

typedef float v2f __attribute__((ext_vector_type(2)));
typedef float v8f __attribute__((ext_vector_type(8)));

// ---------------- zero init (ws poisoned; re-zero every call) -----------
__global__ void lichi_zero(float* __restrict__ Xsum, float* __restrict__ Wsum, int n) {
    int i = blockIdx.x * blockDim.x + threadIdx.x;
    if (i < n) { Xsum[i] = 0.0f; Wsum[i] = 0.0f; }
}

// ---------------- per-position patch squared norms ----------------------
__global__ void lichi_patchsq(const float* __restrict__ y, float* __restrict__ s2g) {
    int i = blockIdx.x * blockDim.x + threadIdx.x;
    if (i >= NS2) return;
    int n = i / (HP * HP), rem = i % (HP * HP);
    int r = rem / HP, c = rem % HP;
    const float* img = y + n * Hh * Ww;
    float s = 0.0f;
    for (int a = 0; a < Pp; ++a) {
        const float* rowp = img + (r + a) * Ww + c;
        #pragma unroll
        for (int b = 0; b < Pp; ++b) {
            float v = rowp[b];
            s = fmaf(v, v, s);
        }
    }
    s2g[i] = s;
}

// ---------------- block matching: WMMA inner products + top-16 ----------
// One workgroup = one image tile of 4x4 reference patches; 8 waves stream
// 16-candidate chunks through V_WMMA_F32_16X16X4_F32 (A=candidates, B=refs^T).
__global__ __launch_bounds__(256) void lichi_bm(const float* __restrict__ y,
                                                const float* __restrict__ s2g,
                                                int* __restrict__ indices) {
    __shared__ float tile[85 * 84];       // union pixel region (slack row for pad chunk)
    __shared__ float listD[256 * 16];     // per-lane private top-16 distances
    __shared__ int   listI[256 * 16];     // per-lane private top-16 cidx
    __shared__ float redD[256];
    __shared__ int   redI[256];

    const int nImg = blockIdx.x / 100;
    const int tIdx = blockIdx.x % 100;
    const int ti = tIdx / 10, tj = tIdx % 10;
    const int Rt = ti * 12, Ct = tj * 12;           // tile ref-pixel origin
    const int tid = threadIdx.x;
    const float* __restrict__ img   = y + nImg * Hh * Ww;
    const float* __restrict__ s2img = s2g + nImg * HP * HP;
    const float FINF = __builtin_inff();

    // stage union region [Rt-32, Rt+51] x [Ct-32, Ct+51] (zeros outside image)
    for (int i = tid; i < TROWS * 84; i += 256) {
        int tr = i / 84, tc = i % 84;
        int gr = Rt - 32 + tr, gc = Ct - 32 + tc;
        float v = 0.0f;
        if (gr >= 0 && gr < Hh && gc >= 0 && gc < Ww) v = img[gr * Ww + gc];
        tile[i] = v;
    }
    for (int i = tid; i < 256 * 16; i += 256) { listD[i] = FINF; listI[i] = 4225; }
    __syncthreads();

    const int wave = tid >> 5;
    const int lane = tid & 31;
    const int half = lane >> 4;
    const int l16  = lane & 15;
    const int ra = l16 >> 2, rb = l16 & 3;          // this lane's ref within tile
    const int refBase = (ra * 3 + 32) * 84 + (rb * 3 + 32);  // tile-local ref patch base
    float* __restrict__ myD = listD + tid * 16;
    int*   __restrict__ myI = listI + tid * 16;
    float curMax = FINF; int curSlot = 0;
    const float inv_pp = 1.0f / (float)NPATCH;

    for (int chunk = wave; chunk < NCHUNK; chunk += 8) {
        // ---- A fragment: candidate patch l16 of this chunk ----
        const int fcA = chunk * 16 + l16;
        const int candBase = (fcA / TGRID) * 84 + (fcA % TGRID);
        v8f cacc = {};
        #pragma unroll
        for (int k0 = 0; k0 < 120; k0 += 4) {       // j <= 119, all valid
            int o0 = half ? OFF(k0 + 2) : OFF(k0 + 0);
            int o1 = half ? OFF(k0 + 3) : OFF(k0 + 1);
            v2f a, b;
            a.x = tile[candBase + o0];  a.y = tile[candBase + o1];
            b.x = tile[refBase  + o0];  b.y = tile[refBase  + o1];
            cacc = __builtin_amdgcn_wmma_f32_16x16x4_f32(
                false, a, false, b, (short)0, cacc, false, false);
        }
        {   // k0 = 120: only K=120 (lanes 0-15) contributes; K=121..123 are pad
            v2f a, b;
            a.x = half ? 0.0f : tile[candBase + OFF(120)];  a.y = 0.0f;
            b.x = half ? 0.0f : tile[refBase  + OFF(120)];  b.y = 0.0f;
            cacc = __builtin_amdgcn_wmma_f32_16x16x4_f32(
                false, a, false, b, (short)0, cacc, false, false);
        }
        // ---- selection: lane holds ip[ref=l16][cand = 8*half + i] ----
        int fcBase = chunk * 16;
        int qrB = fcBase / TGRID, qcB = fcBase % TGRID;
        #pragma unroll
        for (int i = 0; i < 8; ++i) {
            int m = 8 * half + i;
            int qrL = qrB, qcL = qcB + m;
            if (qcL >= TGRID) { qcL -= TGRID; qrL += 1; }
            int dr = qrL - 32 - ra * 3;             // offset relative to this ref
            int dc = qcL - 32 - rb * 3;
            int qr = Rt - 32 + qrL;                 // global candidate top-left
            int qc = Ct - 32 + qcL;
            bool valid = (dr >= -Vv && dr <= Vv && dc >= -Vv && dc <= Vv &&
                          qr >= 0 && qr < HP && qc >= 0 && qc < HP && qrL < TGRID);
            if (valid) {
                float d;
                if (dr == 0 && dc == 0) d = -FINF;  // reference patch always kept
                else d = (s2img[qr * HP + qc] - 2.0f * cacc[i]) * inv_pp;
                if (d < curMax) {
                    myD[curSlot] = d;
                    myI[curSlot] = (dr + Vv) * WINW + (dc + Vv);
                    float mx = myD[0]; int ms = 0;
                    #pragma unroll
                    for (int e = 1; e < 16; ++e) {
                        float v = myD[e];
                        if (v > mx) { mx = v; ms = e; }
                    }
                    curMax = mx; curSlot = ms;
                }
            }
        }
    }
    __syncthreads();

    // ---- merge: 16 lists of 16 per ref -> exact top-16 (tie: smallest cidx) ----
    const int r = tid >> 4, j = tid & 15;
    const int srcTid = (j & 7) * 32 + (j >> 3) * 16 + r;
    float* __restrict__ srcD = listD + srcTid * 16;
    int*   __restrict__ srcI = listI + srcTid * 16;
    const int refRow = ti * 4 + (r >> 2);
    const int refCol = tj * 4 + (r & 3);
    const int gOut = nImg * (HREF * WREF) + refRow * WREF + refCol;

    for (int kSel = 0; kSel < Kk; ++kSel) {
        float bd = FINF; int bpack = 0x7fffffff;
        #pragma unroll
        for (int e = 0; e < 16; ++e) {
            float d = srcD[e];
            int pack = (srcI[e] << 9) | (j << 5) | e;   // cidx major for tie-break
            if (d < bd || (d == bd && pack < bpack)) { bd = d; bpack = pack; }
        }
        redD[tid] = bd; redI[tid] = bpack;
        __syncthreads();
        for (int off = 8; off > 0; off >>= 1) {
            if (j < off) {
                float d2 = redD[tid + off]; int p2 = redI[tid + off];
                if (d2 < redD[tid] || (d2 == redD[tid] && p2 < redI[tid])) {
                    redD[tid] = d2; redI[tid] = p2;
                }
            }
            __syncthreads();
        }
        int winPack = redI[r * 16];
        if (j == 0) {
            int cidx = winPack >> 9;
            int dr = cidx / WINW - Vv, dc = cidx % WINW - Vv;
            indices[gOut * Kk + kSel] = (refRow * 3 + dr) * HP + (refCol * 3 + dc);
        }
        if (j == ((winPack >> 5) & 15)) srcD[winPack & 15] = FINF;
        __syncthreads();
    }
}

// ---------------- per-group denoise: Gram(WMMA) + Cholesky + theta(WMMA) ---
__global__ __launch_bounds__(32) void lichi_denoise(const float* __restrict__ y,
                                                    const int* __restrict__ indices,
                                                    float* __restrict__ Xsum,
                                                    float* __restrict__ Wsum) {
    __shared__ float Ys[Kk][KPAD];
    __shared__ float Qs[Kk][Kk + 1];
    __shared__ float Qi[Kk][Kk + 1];
    __shared__ float Ts[Kk][Kk + 1];
    __shared__ float q1[Kk];
    __shared__ float wgt[Kk];
    __shared__ int   rtl[Kk], ctl[Kk];

    const int g = blockIdx.x;
    const int n = g / (HREF * WREF);
    const float* __restrict__ img = y + n * Hh * Ww;
    const int lane = threadIdx.x;
    const int half = lane >> 4;
    const int l16  = lane & 15;

    if (lane < Kk) {
        int ind = indices[g * Kk + lane];
        rtl[lane] = ind / HP;
        ctl[lane] = ind % HP;
    }
    __syncthreads();

    for (int tIdx = lane; tIdx < Kk * KPAD; tIdx += 32) {
        int m = tIdx >> 7;
        int j = tIdx & 127;
        float v = 0.0f;
        if (j < NPATCH)
            v = img[(rtl[m] + j / Pp) * Ww + (ctl[m] + j % Pp)];
        Ys[m][j] = v;
    }
    __syncthreads();

    // Q = Y * Y^T + E*I (A and B fragments identical for the Gram)
    v8f qacc = {};
    for (int k0 = 0; k0 < KPAD; k0 += 4) {
        v2f a;
        a.x = Ys[l16][k0 + 2 * half + 0];
        a.y = Ys[l16][k0 + 2 * half + 1];
        qacc = __builtin_amdgcn_wmma_f32_16x16x4_f32(
            false, a, false, a, (short)0, qacc, false, false);
    }
    #pragma unroll
    for (int i = 0; i < 8; ++i) {
        int row = i + 8 * half;
        float v = qacc[i];
        if (row == l16) v += E_CONST;
        Qs[row][l16] = v;
    }
    __syncthreads();

    // Cholesky (in place, lower), 16 lanes each own a row
    for (int kcol = 0; kcol < Kk; ++kcol) {
        if (lane == kcol) Qs[kcol][kcol] = sqrtf(Qs[kcol][kcol]);
        __syncthreads();
        if (lane < Kk && lane > kcol) Qs[lane][kcol] /= Qs[kcol][kcol];
        __syncthreads();
        if (lane < Kk && lane > kcol) {
            float lik = Qs[lane][kcol];
            for (int j = kcol + 1; j <= lane; ++j)
                Qs[lane][j] -= lik * Qs[j][kcol];
        }
        __syncthreads();
    }

    // Q^-1: each of 16 lanes solves L L^T x = e_c for its column
    if (lane < Kk) {
        const int c = lane;
        float z[Kk], x[Kk];
        #pragma unroll
        for (int i = 0; i < Kk; ++i) {
            float s = (i == c) ? 1.0f : 0.0f;
            #pragma unroll
            for (int j = 0; j < Kk; ++j)
                if (j < i) s -= Qs[i][j] * z[j];
            z[i] = s / Qs[i][i];
        }
        #pragma unroll
        for (int i = Kk - 1; i >= 0; --i) {
            float s = z[i];
            #pragma unroll
            for (int j = 0; j < Kk; ++j)
                if (j > i) s -= Qs[j][i] * x[j];
            x[i] = s / Qs[i][i];
        }
        #pragma unroll
        for (int i = 0; i < Kk; ++i) Qi[i][c] = x[i];
    }
    __syncthreads();

    if (lane < Kk) {
        float s = 0.0f;
        #pragma unroll
        for (int j = 0; j < Kk; ++j) s += Qi[lane][j];
        q1[lane] = s;
    }
    __syncthreads();
    float q2 = 0.0f;
    #pragma unroll
    for (int j = 0; j < Kk; ++j) q2 += q1[j];
    if (lane < Kk) {
        float wsum = 0.0f;
        #pragma unroll
        for (int j = 0; j < Kk; ++j) {
            float tv = ((lane == j) ? 1.0f : 0.0f)
                       - (Qi[lane][j] - q1[lane] * q1[j] / q2) * DE_CONST;
            Ts[lane][j] = tv;
            wsum = fmaf(tv, tv, wsum);
        }
        float cl = fminf(fmaxf(wsum, 1.0f / (float)Kk), 1.0f);
        wgt[lane] = 1.0f / cl;
    }
    __syncthreads();

    // X_hat = theta * Y via WMMA; scatter weighted patches + weights
    float* __restrict__ xbase = Xsum + n * Hh * Ww;
    float* __restrict__ wbase = Wsum + n * Hh * Ww;
    for (int b = 0; b < 8; ++b) {
        v8f cacc = {};
        #pragma unroll
        for (int k0 = 0; k0 < Kk; k0 += 4) {
            v2f a, bb;
            a.x  = Ts[l16][k0 + 2 * half + 0];
            a.y  = Ts[l16][k0 + 2 * half + 1];
            bb.x = Ys[k0 + 2 * half + 0][b * 16 + l16];
            bb.y = Ys[k0 + 2 * half + 1][b * 16 + l16];
            cacc = __builtin_amdgcn_wmma_f32_16x16x4_f32(
                false, a, false, bb, (short)0, cacc, false, false);
        }
        int j = b * 16 + l16;
        if (j < NPATCH) {
            int pa = j / Pp, pb = j % Pp;
            #pragma unroll
            for (int i = 0; i < 8; ++i) {
                int m = i + 8 * half;
                float w = wgt[m];
                int off = (rtl[m] + pa) * Ww + (ctl[m] + pb);
                atomicAdd(xbase + off, cacc[i] * w);
                atomicAdd(wbase + off, w);
            }
        }
    }
}

// ---------------- final divide -------------------------------------------
__global__ void lichi_div(const float* __restrict__ Xsum,
                          const float* __restrict__ Wsum,
                          float* __restrict__ out, int n) {
    int i = blockIdx.x * blockDim.x + threadIdx.x;
    if (i < n) out[i] = Xsum[i] / Wsum[i];
}

extern "C" void kernel_launch(void* const* d_in, const int* in_sizes, int n_in,
                              void* d_out, int out_size, void* d_ws, size_t ws_size,
                              hipStream_t stream) {
    const float* y = (const float*)d_in[0];     // (4,1,128,128) f32
    float* out = (float*)d_out;                 // (4,1,128,128) f32

    char* ws = (char*)d_ws;
    int*   indices = (int*)ws;                                        // 409600 B
    float* Xsum = (float*)(ws + (size_t)NGRP * Kk * sizeof(int));     // 262144 B
    float* Wsum = Xsum + NPIX;                                        // 262144 B
    float* s2g  = Wsum + NPIX;                                        // 222784 B

    lichi_zero<<<(NPIX + 255) / 256, 256, 0, stream>>>(Xsum, Wsum, NPIX);
    lichi_patchsq<<<(NS2 + 255) / 256, 256, 0, stream>>>(y, s2g);
    lichi_bm<<<400, 256, 0, stream>>>(y, s2g, indices);
    lichi_denoise<<<NGRP, 32, 0, stream>>>(y, indices, Xsum, Wsum);
    lichi_div<<<(NPIX + 255) / 256, 256, 0, stream>>>(Xsum, Wsum, out, NPIX);
}